// Tagger_53833120088739
// MI455X (gfx1250) — compile-verified
//
#include <hip/hip_runtime.h>
#include <hip/hip_bf16.h>

typedef __attribute__((ext_vector_type(2))) float v2f;
typedef __attribute__((ext_vector_type(8))) float v8f;

#define S_LEN 4096
#define B_SZ  64
#define E_RAW 50
#define E_PAD 64
#define T_TAG 20

// ---------------------------------------------------------------------------
// 1) Embedding gather into channel-minor layout x[B][S][E_PAD]
//    (channels >= 50 zeroed). Fully coalesced writes, no staging needed.
//    Block 256 = 4 tokens x 64 channels. Grid (S/4, B).
// ---------------------------------------------------------------------------
__global__ __launch_bounds__(256) void embed_kernel(
    const int* __restrict__ token_id, const float* __restrict__ table,
    float* __restrict__ x) {
  const int b = blockIdx.y;
  const int s = blockIdx.x * 4 + (threadIdx.x >> 6);
  const int e = threadIdx.x & 63;
  const int tok = token_id[(size_t)b * S_LEN + s];
  const float v = (e < E_RAW) ? table[(size_t)tok * E_RAW + e] : 0.f;
  x[((size_t)b * S_LEN + s) * E_PAD + e] = v;
}

// ---------------------------------------------------------------------------
// 2) Conv1d(k=3, SAME) as WMMA f32 GEMM + fused swish residual.
//    Activations channel-minor: x[B][S][E_PAD]. Block = 256 threads (8 waves)
//    computes y[b][s0:s0+32][0:64]. Wave tile 16(M=e_out) x 16(N=s);
//    3 taps x 16 K-steps of v_wmma_f32_16x16x4_f32 = 48 WMMA / wave.
//    A and B fragments are both contiguous 8B LDS reads -> ds_load_2addr_b64.
//    Epilogue: each lane owns 8 consecutive channels of one s row ->
//    2x ds_load_b128 (residual) + 2x global_store_b128.
// ---------------------------------------------------------------------------
__global__ __launch_bounds__(256) void conv_swish_kernel(
    const float* __restrict__ xin, float* __restrict__ xout,
    const float* __restrict__ conv_w, const float* __restrict__ conv_b) {
  __shared__ float Wlds[3][E_PAD][68];   // [k][e_out][e_in], stride 68: no conflicts
  __shared__ float slab[34][68];         // [j][e], j = s - s0 + 1 (halo rows)
  __shared__ float blds[E_PAD];          // zero-padded bias

  const int b   = blockIdx.y;
  const int s0  = blockIdx.x * 32;
  const int tid = threadIdx.x;

  // Stage padded weights
  for (int i = tid; i < 3 * E_PAD * E_PAD; i += 256) {
    const int k = i >> 12;
    const int r = i & 4095;
    const int eo = r >> 6, ei = r & 63;
    float w = 0.f;
    if (eo < E_RAW && ei < E_RAW) w = conv_w[((size_t)eo * E_RAW + ei) * 3 + k];
    Wlds[k][eo][ei] = w;
  }
  // Stage input slab (straight coalesced copy of 34 channel rows + halo pad)
  const float* xb = xin + (size_t)b * S_LEN * E_PAD;
  for (int i = tid; i < 34 * E_PAD; i += 256) {
    const int j = i >> 6, e = i & 63;
    const int s = s0 - 1 + j;
    slab[j][e] = (s >= 0 && s < S_LEN) ? xb[(size_t)s * E_PAD + e] : 0.f;
  }
  if (tid < E_PAD) blds[tid] = (tid < E_RAW) ? conv_b[tid] : 0.f;
  __syncthreads();

  const int wave  = tid >> 5;
  const int lane  = tid & 31;
  const int l16   = lane & 15;
  const int khalf = (lane < 16) ? 0 : 2;   // K split across lane halves
  const int m0    = (wave & 3) * 16;       // e_out tile
  const int n0    = (wave >> 2) * 16;      // s tile within block

  v8f c = {};
  #pragma unroll
  for (int k = 0; k < 3; ++k) {
    const int brow = n0 + l16 + k;         // slab row for this tap
    #pragma unroll
    for (int e4 = 0; e4 < 16; ++e4) {
      const int e0 = e4 * 4 + khalf;
      v2f a, bm;
      a.x  = Wlds[k][m0 + l16][e0];
      a.y  = Wlds[k][m0 + l16][e0 + 1];
      bm.x = slab[brow][e0];
      bm.y = slab[brow][e0 + 1];
      c = __builtin_amdgcn_wmma_f32_16x16x4_f32(
              /*neg_a=*/false, a, /*neg_b=*/false, bm,
              /*c_mod=*/(short)0, c, /*reuse_a=*/false, /*reuse_b=*/false);
    }
  }

  // Epilogue: bias + swish + residual. Lane owns channels [eo0, eo0+8) of
  // sequence position s0 + sl -> contiguous 32B stores.
  const int hi  = (lane < 16) ? 0 : 8;
  const int sl  = n0 + l16;
  const int eo0 = m0 + hi;
  float* orow = xout + ((size_t)b * S_LEN + (s0 + sl)) * E_PAD + eo0;
  #pragma unroll
  for (int r = 0; r < 8; ++r) {
    const float y  = c[r] + blds[eo0 + r];
    const float sw = y * (1.f / (1.f + __expf(-y)));
    orow[r] = sw + slab[sl + 1][eo0 + r];
  }
}

// ---------------------------------------------------------------------------
// 3) Emissions: em[s][b][t] = out_w[t,:].x[b,s,:] + out_b[t]  (T=20, tiny)
//    x channel-minor -> each thread reads its 64 padded channels as float4s.
// ---------------------------------------------------------------------------
__global__ __launch_bounds__(256) void emissions_kernel(
    const float* __restrict__ x, const float* __restrict__ out_w,
    const float* __restrict__ out_b, float* __restrict__ em) {
  __shared__ float wl[T_TAG][E_PAD];     // zero-padded weights
  __shared__ float bl[T_TAG];
  const int b = blockIdx.y;
  const int s = blockIdx.x * 256 + threadIdx.x;
  for (int i = threadIdx.x; i < T_TAG * E_PAD; i += 256) {
    const int t = i >> 6, e = i & 63;
    wl[t][e] = (e < E_RAW) ? out_w[t * E_RAW + e] : 0.f;
  }
  if (threadIdx.x < T_TAG) bl[threadIdx.x] = out_b[threadIdx.x];
  __syncthreads();

  float acc[T_TAG];
  #pragma unroll
  for (int t = 0; t < T_TAG; ++t) acc[t] = bl[t];

  const float* xr = x + ((size_t)b * S_LEN + s) * E_PAD;
  #pragma unroll
  for (int e4 = 0; e4 < E_PAD / 4; ++e4) {   // padded channels are zero
    const float4 xv = *(const float4*)(xr + e4 * 4);
    #pragma unroll
    for (int t = 0; t < T_TAG; ++t) {
      acc[t] += wl[t][e4 * 4 + 0] * xv.x + wl[t][e4 * 4 + 1] * xv.y +
                wl[t][e4 * 4 + 2] * xv.z + wl[t][e4 * 4 + 3] * xv.w;
    }
  }
  float* o = em + ((size_t)s * B_SZ + b) * T_TAG;
  #pragma unroll
  for (int t = 0; t < T_TAG; ++t) o[t] = acc[t];
}

// ---------------------------------------------------------------------------
// 4) CRF forward algorithm (denominator). One wave32 per batch element;
//    lane = tag. Transition column in registers, alpha broadcast via shfl.
// ---------------------------------------------------------------------------
__global__ __launch_bounds__(32) void crf_forward_kernel(
    const float* __restrict__ em, const float* __restrict__ start_trans,
    const float* __restrict__ end_trans, const float* __restrict__ trans,
    float* __restrict__ den) {
  const int b = blockIdx.x;
  const int t = threadIdx.x;
  const bool act = (t < T_TAG);
  const int tt = act ? t : 0;
  float tr[T_TAG];
  #pragma unroll
  for (int i = 0; i < T_TAG; ++i) tr[i] = trans[i * T_TAG + tt];

  float alpha = act ? (start_trans[tt] + em[(size_t)b * T_TAG + tt]) : -1e30f;
  for (int s = 1; s < S_LEN; ++s) {
    float v[T_TAG];
    float m = -1e30f;
    #pragma unroll
    for (int i = 0; i < T_TAG; ++i) {
      v[i] = __shfl(alpha, i, 32) + tr[i];
      m = fmaxf(m, v[i]);
    }
    float sum = 0.f;
    #pragma unroll
    for (int i = 0; i < T_TAG; ++i) sum += expf(v[i] - m);
    const float e_t = act ? em[((size_t)s * B_SZ + b) * T_TAG + tt] : 0.f;
    alpha = act ? (m + logf(sum) + e_t) : -1e30f;
  }
  float vv = act ? (alpha + end_trans[tt]) : -1e30f;
  float m = vv;
  for (int off = 16; off > 0; off >>= 1) m = fmaxf(m, __shfl_xor(m, off, 32));
  float ex = act ? expf(vv - m) : 0.f;
  for (int off = 16; off > 0; off >>= 1) ex += __shfl_xor(ex, off, 32);
  if (t == 0) den[b] = m + logf(ex);
}

// ---------------------------------------------------------------------------
// 5) Gold-path score (numerator). Block per batch element.
// ---------------------------------------------------------------------------
__global__ __launch_bounds__(256) void crf_numerator_kernel(
    const float* __restrict__ em, const int* __restrict__ tag_id,
    const float* __restrict__ start_trans, const float* __restrict__ end_trans,
    const float* __restrict__ trans, float* __restrict__ num) {
  __shared__ float red[256];
  const int b = blockIdx.x;
  const int tid = threadIdx.x;
  const int* tg = tag_id + (size_t)b * S_LEN;
  float acc = 0.f;
  for (int s = tid; s < S_LEN; s += 256) {
    const int t = tg[s];
    acc += em[((size_t)s * B_SZ + b) * T_TAG + t];
    if (s + 1 < S_LEN) acc += trans[t * T_TAG + tg[s + 1]];
  }
  if (tid == 0) acc += start_trans[tg[0]] + end_trans[tg[S_LEN - 1]];
  red[tid] = acc;
  __syncthreads();
  for (int w = 128; w > 0; w >>= 1) {
    if (tid < w) red[tid] += red[tid + w];
    __syncthreads();
  }
  if (tid == 0) num[b] = red[0];
}

// ---------------------------------------------------------------------------
// 6) -llh = sum_b (den[b] - num[b])
// ---------------------------------------------------------------------------
__global__ __launch_bounds__(64) void final_reduce_kernel(
    const float* __restrict__ den, const float* __restrict__ num,
    float* __restrict__ out) {
  __shared__ float red[64];
  const int tid = threadIdx.x;
  red[tid] = den[tid] - num[tid];
  __syncthreads();
  for (int w = 32; w > 0; w >>= 1) {
    if (tid < w) red[tid] += red[tid + w];
    __syncthreads();
  }
  if (tid == 0) out[0] = red[0];
}

extern "C" void kernel_launch(void* const* d_in, const int* in_sizes, int n_in,
                              void* d_out, int out_size, void* d_ws, size_t ws_size,
                              hipStream_t stream) {
  const int*   token_id    = (const int*)d_in[0];
  const int*   tag_id      = (const int*)d_in[1];
  const float* embed_table = (const float*)d_in[2];
  const float* conv_w      = (const float*)d_in[3];
  const float* conv_b      = (const float*)d_in[4];
  const float* out_w       = (const float*)d_in[5];
  const float* out_b       = (const float*)d_in[6];
  const float* start_trans = (const float*)d_in[7];
  const float* end_trans   = (const float*)d_in[8];
  const float* trans       = (const float*)d_in[9];
  float* out = (float*)d_out;

  char* ws = (char*)d_ws;
  const size_t XBYTES  = (size_t)B_SZ * S_LEN * E_PAD * sizeof(float);   // 64 MB
  const size_t EMBYTES = (size_t)S_LEN * B_SZ * T_TAG * sizeof(float);   // 21 MB
  float* x0  = (float*)(ws);
  float* x1  = (float*)(ws + XBYTES);
  float* em  = (float*)(ws + 2 * XBYTES);
  float* den = (float*)(ws + 2 * XBYTES + EMBYTES);
  float* num = den + B_SZ;

  // 1) embed -> x0 [B][S][E_PAD]
  embed_kernel<<<dim3(S_LEN / 4, B_SZ), 256, 0, stream>>>(token_id, embed_table, x0);

  // 2) 5x conv + swish residual, ping-pong x0 <-> x1
  const float* cin = x0;
  float* cout = x1;
  for (int it = 0; it < 5; ++it) {
    conv_swish_kernel<<<dim3(S_LEN / 32, B_SZ), 256, 0, stream>>>(cin, cout, conv_w, conv_b);
    float* tmp = (float*)cin;
    cin = cout;
    cout = tmp;
  }

  // 3) emissions from latest buffer (cin)
  emissions_kernel<<<dim3(S_LEN / 256, B_SZ), 256, 0, stream>>>(cin, out_w, out_b, em);

  // 4) CRF forward (denominator), 5) numerator, 6) final sum
  crf_forward_kernel<<<B_SZ, 32, 0, stream>>>(em, start_trans, end_trans, trans, den);
  crf_numerator_kernel<<<B_SZ, 256, 0, stream>>>(em, tag_id, start_trans, end_trans, trans, num);
  final_reduce_kernel<<<1, 64, 0, stream>>>(den, num, out);
}